// FinePreprocess_77094662963352
// MI455X (gfx1250) — compile-verified
//
#include <hip/hip_runtime.h>
#include <hip/hip_bf16.h>

typedef float v2f __attribute__((ext_vector_type(2)));
typedef float v8f __attribute__((ext_vector_type(8)));

#define BB   4
#define FF   64
#define HH   256
#define WWD  256
#define HW   (HH * WWD)      // 65536
#define WPAD (WWD + 6)       // 262
#define NPIX ((size_t)BB * HW)

// ---------------------------------------------------------------------------
// K0: positional-encoding MLP (49 x 2 -> ... -> 49 x 64), one tiny block.
// ---------------------------------------------------------------------------
__device__ inline void dense_layer(const float* w, const float* b,
                                   const float* x, float* y,
                                   int din, int dout, bool relu) {
  for (int j = 0; j < dout; ++j) {
    float acc = b[j];
    for (int f = 0; f < din; ++f) acc += w[j * din + f] * x[f];
    y[j] = relu ? fmaxf(acc, 0.f) : acc;
  }
}

__global__ void k_pe(const float* w0, const float* b0, const float* w1, const float* b1,
                     const float* w2, const float* b2, const float* w3, const float* b3,
                     const float* w4, const float* b4, float* pe) {
  int t = threadIdx.x;
  if (t >= 49) return;
  float xa[64], xb[64];
  xa[0] = (float)(t % 7 - 3);   // cols offset (varies fastest)
  xa[1] = (float)(t / 7 - 3);   // rows offset
  dense_layer(w0, b0, xa, xb, 2, 8, true);
  dense_layer(w1, b1, xb, xa, 8, 16, true);
  dense_layer(w2, b2, xa, xb, 16, 32, true);
  dense_layer(w3, b3, xb, xa, 32, 64, true);
  dense_layer(w4, b4, xa, xb, 64, 64, false);
  for (int j = 0; j < 64; ++j) pe[t * 64 + j] = xb[j];
}

// ---------------------------------------------------------------------------
// K1: fused transpose (B,F,H,W)->(pix,F) + kv projection GEMM via WMMA f32.
// One wave per 16-pixel x 64-channel tile. wkv staged in LDS (padded stride).
// ---------------------------------------------------------------------------
__global__ void k_proj_transpose(const float* __restrict__ f1,
                                 const float* __restrict__ wkv,
                                 float* __restrict__ f1t,
                                 float* __restrict__ kvt) {
  __shared__ float wlds[64 * 66];
  for (int i = threadIdx.x; i < 64 * 64; i += blockDim.x)
    wlds[(i >> 6) * 66 + (i & 63)] = wkv[i];
  __syncthreads();

  int lane = threadIdx.x & 31;
  int tile = blockIdx.x * (blockDim.x >> 5) + (threadIdx.x >> 5);
  int p0   = tile * 16;                    // global pixel base (b-major)
  int img  = p0 >> 16;                     // /65536
  int pix  = p0 & 65535;
  const float* src = f1 + (size_t)img * FF * HW;
  int m  = lane & 15;
  int hf = lane >> 4;

  v8f acc[4] = {};
  for (int k0 = 0; k0 < 64; k0 += 4) {
    int k = k0 + 2 * hf;
    v2f a;
    a.x = src[(size_t)k * HW + pix + m];
    a.y = src[(size_t)(k + 1) * HW + pix + m];
    // write raw transposed tile (row-major pixel x channel)
    *(v2f*)(f1t + (size_t)(p0 + m) * 64 + k) = a;
#pragma unroll
    for (int t = 0; t < 4; ++t) {
      const float* wp = &wlds[(t * 16 + m) * 66 + k];
      v2f bf; bf.x = wp[0]; bf.y = wp[1];
      acc[t] = __builtin_amdgcn_wmma_f32_16x16x4_f32(
          false, a, false, bf, (short)0, acc[t], false, false);
    }
  }
#pragma unroll
  for (int t = 0; t < 4; ++t)
#pragma unroll
    for (int i = 0; i < 8; ++i) {
      int mm = i + 8 * hf;
      kvt[(size_t)(p0 + mm) * 64 + t * 16 + m] = acc[t][i];
    }
}

// ---------------------------------------------------------------------------
// K2: gather query0 (center (2,2) of each 4x4 block) from channel-major f0.
// ---------------------------------------------------------------------------
__global__ void k_gather_q0(const float* __restrict__ f0,
                            const int* __restrict__ b_ids,
                            const int* __restrict__ i_ids,
                            float* __restrict__ q0, int N) {
  int idx = blockIdx.x * blockDim.x + threadIdx.x;
  int n = idx >> 6, f = idx & 63;
  if (n >= N) return;
  int b = b_ids[n], i = i_ids[n];
  int r = (i >> 6) * 4 + 2;     // w/4 == 64
  int c = (i & 63) * 4 + 2;
  q0[(size_t)n * 64 + f] = f0[(((size_t)b * FF + f) * HH + r) * WWD + c];
}

// ---------------------------------------------------------------------------
// K3: generic C(Mx64) = A(MxK) @ W(64xK)^T via WMMA f32 16x16x4.
// Optional: concat A (A1 supplies k>=64), relu, residual add.
// ---------------------------------------------------------------------------
__global__ void k_gemm64(const float* __restrict__ A0, const float* __restrict__ A1,
                         const float* __restrict__ W, float* __restrict__ C,
                         const float* __restrict__ resid, int M, int K, int relu) {
  __shared__ float wlds[64 * 130];
  int stride = K + 2;
  for (int i = threadIdx.x; i < 64 * K; i += blockDim.x)
    wlds[(i / K) * stride + (i % K)] = W[i];
  __syncthreads();

  int lane = threadIdx.x & 31;
  int tile = blockIdx.x * (blockDim.x >> 5) + (threadIdx.x >> 5);
  int row0 = tile * 16;
  if (row0 >= M) return;
  int m  = lane & 15;
  int hf = lane >> 4;

  v8f acc[4] = {};
  for (int k0 = 0; k0 < K; k0 += 4) {
    int k = k0 + 2 * hf;
    const float* ap = (A1 && k >= 64)
                          ? (A1 + (size_t)(row0 + m) * 64 + (k - 64))
                          : (A0 + (size_t)(row0 + m) * 64 + k);
    v2f a = *(const v2f*)ap;
#pragma unroll
    for (int t = 0; t < 4; ++t) {
      const float* wp = &wlds[(t * 16 + m) * stride + k];
      v2f bf; bf.x = wp[0]; bf.y = wp[1];
      acc[t] = __builtin_amdgcn_wmma_f32_16x16x4_f32(
          false, a, false, bf, (short)0, acc[t], false, false);
    }
  }
#pragma unroll
  for (int t = 0; t < 4; ++t)
#pragma unroll
    for (int i = 0; i < 8; ++i) {
      int mm = i + 8 * hf;
      size_t off = (size_t)(row0 + mm) * 64 + t * 16 + m;
      float v = acc[t][i];
      if (relu) v = fmaxf(v, 0.f);
      if (resid) v += resid[off];
      C[off] = v;
    }
}

// ---------------------------------------------------------------------------
// K4: per-query windowed attention (4 heads x 16 dims over 49 keys).
// One wave per query; keys gathered from L2-resident kvt (+pe), zero-padded.
// ---------------------------------------------------------------------------
__global__ void k_attn(const float* __restrict__ q2, const float* __restrict__ kvt,
                       const float* __restrict__ pe, const int* __restrict__ b_ids,
                       const int* __restrict__ j_ids, float* __restrict__ ao, int N) {
  __shared__ float sc[8][200];
  int lane = threadIdx.x & 31;
  int wv   = threadIdx.x >> 5;
  int nq   = blockIdx.x * (blockDim.x >> 5) + wv;
  bool valid = nq < N;
  int n = valid ? nq : 0;

  int b = b_ids[n], j = j_ids[n];
  int r = j / WPAD, c = j % WPAD;          // padded coords, center in-bounds
  float qa = q2[(size_t)n * 64 + lane];
  float qb = q2[(size_t)n * 64 + lane + 32];
  const float* kb = kvt + (size_t)b * HW * 64;
  float* mys = sc[wv];

  // pass 1: scores
  for (int s = 0; s < 49; ++s) {
    int ur = r + s / 7 - 6, uc = c + s % 7 - 6;
    bool inb = (ur >= 0) & (ur < HH) & (uc >= 0) & (uc < WWD);
    const float* kp = kb + ((size_t)ur * WWD + uc) * 64;
    float k0v = (inb ? kp[lane]      : 0.f) + pe[s * 64 + lane];
    float k1v = (inb ? kp[lane + 32] : 0.f) + pe[s * 64 + lane + 32];
    float p0 = qa * k0v, p1 = qb * k1v;
#pragma unroll
    for (int o = 8; o >= 1; o >>= 1) {     // reduce within 16-lane head groups
      p0 += __shfl_xor(p0, o, 32);
      p1 += __shfl_xor(p1, o, 32);
    }
    if (lane == 0)  { mys[s * 4 + 0] = p0 * 0.25f; mys[s * 4 + 2] = p1 * 0.25f; }
    if (lane == 16) { mys[s * 4 + 1] = p0 * 0.25f; mys[s * 4 + 3] = p1 * 0.25f; }
  }
  __syncthreads();
  // softmax per head (lanes 0..3)
  if (lane < 4) {
    float mx = -1e30f;
    for (int s = 0; s < 49; ++s) mx = fmaxf(mx, mys[s * 4 + lane]);
    float sum = 0.f;
    for (int s = 0; s < 49; ++s) {
      float e = __expf(mys[s * 4 + lane] - mx);
      mys[s * 4 + lane] = e; sum += e;
    }
    float inv = 1.f / sum;
    for (int s = 0; s < 49; ++s) mys[s * 4 + lane] *= inv;
  }
  __syncthreads();
  // pass 2: weighted sum of values (== keys)
  float o0 = 0.f, o1 = 0.f;
  int h0 = lane >> 4, h1i = 2 + (lane >> 4);
  for (int s = 0; s < 49; ++s) {
    int ur = r + s / 7 - 6, uc = c + s % 7 - 6;
    bool inb = (ur >= 0) & (ur < HH) & (uc >= 0) & (uc < WWD);
    const float* kp = kb + ((size_t)ur * WWD + uc) * 64;
    float k0v = (inb ? kp[lane]      : 0.f) + pe[s * 64 + lane];
    float k1v = (inb ? kp[lane + 32] : 0.f) + pe[s * 64 + lane + 32];
    o0 += mys[s * 4 + h0]  * k0v;
    o1 += mys[s * 4 + h1i] * k1v;
  }
  if (valid) {
    ao[(size_t)n * 64 + lane]      = o0;
    ao[(size_t)n * 64 + lane + 32] = o1;
  }
}

// ---------------------------------------------------------------------------
// K5: confidence column: sigmoid(h1 . wmlp2[64,:]). One wave per query.
// ---------------------------------------------------------------------------
__global__ void k_conf(const float* __restrict__ h1, const float* __restrict__ wrow,
                       float* __restrict__ conf, int N) {
  int lane = threadIdx.x & 31;
  int n = blockIdx.x * (blockDim.x >> 5) + (threadIdx.x >> 5);
  if (n >= N) return;
  float p = h1[(size_t)n * 64 + lane] * wrow[lane] +
            h1[(size_t)n * 64 + lane + 32] * wrow[lane + 32];
#pragma unroll
  for (int o = 16; o >= 1; o >>= 1) p += __shfl_xor(p, o, 32);
  if (lane == 0) conf[n] = 1.f / (1.f + __expf(-p));
}

// ---------------------------------------------------------------------------
// K6: feat1 = gather(f1t windows, zero-padded) + pe. One wave per (n,s).
// ---------------------------------------------------------------------------
__global__ void k_feat1(const float* __restrict__ f1t, const float* __restrict__ pe,
                        const int* __restrict__ b_ids, const int* __restrict__ j_ids,
                        float* __restrict__ out, int N) {
  int lane = threadIdx.x & 31;
  int idx = blockIdx.x * (blockDim.x >> 5) + (threadIdx.x >> 5);
  if (idx >= N * 49) return;
  int n = idx / 49, s = idx % 49;
  int b = b_ids[n], j = j_ids[n];
  int r = j / WPAD, c = j % WPAD;
  int ur = r + s / 7 - 6, uc = c + s % 7 - 6;
  bool inb = (ur >= 0) & (ur < HH) & (uc >= 0) & (uc < WWD);
  v2f v = {};
  if (inb)
    v = *(const v2f*)(f1t + ((size_t)b * HW + (size_t)ur * WWD + uc) * 64 + 2 * lane);
  v2f p = *(const v2f*)(pe + s * 64 + 2 * lane);
  v.x += p.x; v.y += p.y;
  *(v2f*)(out + ((size_t)n * 49 + s) * 64 + 2 * lane) = v;
}

// ---------------------------------------------------------------------------
extern "C" void kernel_launch(void* const* d_in, const int* in_sizes, int n_in,
                              void* d_out, int out_size, void* d_ws, size_t ws_size,
                              hipStream_t stream) {
  const float* feat_f0 = (const float*)d_in[0];
  const float* feat_f1 = (const float*)d_in[1];
  const int* b_ids = (const int*)d_in[2];
  const int* i_ids = (const int*)d_in[3];
  const int* j_ids = (const int*)d_in[4];
  const float* kw0 = (const float*)d_in[5];  const float* kb0 = (const float*)d_in[6];
  const float* kw1 = (const float*)d_in[7];  const float* kb1 = (const float*)d_in[8];
  const float* kw2 = (const float*)d_in[9];  const float* kb2 = (const float*)d_in[10];
  const float* kw3 = (const float*)d_in[11]; const float* kb3 = (const float*)d_in[12];
  const float* kw4 = (const float*)d_in[13]; const float* kb4 = (const float*)d_in[14];
  const float* wq     = (const float*)d_in[15];
  const float* wkv    = (const float*)d_in[16];
  const float* wmerge = (const float*)d_in[17];
  const float* wmlp1  = (const float*)d_in[18];
  const float* wmlp2  = (const float*)d_in[19];
  int N = in_sizes[2];

  // workspace layout (floats)
  float* ws  = (float*)d_ws;
  float* f1t = ws;                         // NPIX*64
  float* kvt = f1t + NPIX * 64;            // NPIX*64
  float* pe  = kvt + NPIX * 64;            // 49*64 (padded to 4096)
  float* q0  = pe + 4096;                  // N*64
  float* q2  = q0 + (size_t)N * 64;
  float* ao  = q2 + (size_t)N * 64;
  float* mg  = ao + (size_t)N * 64;
  float* h1  = mg + (size_t)N * 64;

  float* out_feat0 = (float*)d_out;                         // N*64
  float* out_feat1 = out_feat0 + (size_t)N * 64;            // N*49*64
  float* out_conf  = out_feat1 + (size_t)N * 49 * 64;       // N

  int gemm_blocks = ((N + 15) / 16 + 7) / 8;

  k_pe<<<1, 64, 0, stream>>>(kw0, kb0, kw1, kb1, kw2, kb2, kw3, kb3, kw4, kb4, pe);

  int proj_blocks = (int)(NPIX / 16 / 8);   // 2048
  k_proj_transpose<<<proj_blocks, 256, 0, stream>>>(feat_f1, wkv, f1t, kvt);

  k_gather_q0<<<(N * 64 + 255) / 256, 256, 0, stream>>>(feat_f0, b_ids, i_ids, q0, N);

  // q2 = q0 @ wq^T
  k_gemm64<<<gemm_blocks, 256, 0, stream>>>(q0, nullptr, wq, q2, nullptr, N, 64, 0);

  k_attn<<<(N + 7) / 8, 256, 0, stream>>>(q2, kvt, pe, b_ids, j_ids, ao, N);

  // merged = ao @ wmerge^T
  k_gemm64<<<gemm_blocks, 256, 0, stream>>>(ao, nullptr, wmerge, mg, nullptr, N, 64, 0);

  // h1 = relu([q0 | merged] @ wmlp1^T)
  k_gemm64<<<gemm_blocks, 256, 0, stream>>>(q0, mg, wmlp1, h1, nullptr, N, 128, 1);

  // feat0 = h1 @ wmlp2[0:64]^T + q0
  k_gemm64<<<gemm_blocks, 256, 0, stream>>>(h1, nullptr, wmlp2, out_feat0, q0, N, 64, 0);

  k_conf<<<(N + 7) / 8, 256, 0, stream>>>(h1, wmlp2 + 64 * 64, out_conf, N);

  k_feat1<<<(N * 49 + 7) / 8, 256, 0, stream>>>(f1t, pe, b_ids, j_ids, out_feat1, N);
}